// SelfAttention_40475771798071
// MI455X (gfx1250) — compile-verified
//
#include <hip/hip_runtime.h>

// ---------------- problem constants ----------------
#define B_    8
#define CIN   256
#define H_    56
#define W_    56
#define HW    3136          // 56*56
#define REL   64
#define OUTC  256
#define SHARE 8
#define KSZ   3
#define KK    9             // KSZ*KSZ
#define MID   32            // OUTC/SHARE
#define CCAT  640           // REL*(1+KK)
#define EPS   1e-3f
#define OC1   384           // fused proj rows: 64(w1)+64(w2)+256(w3)
#define NTILES_P (HW/16)    // 196 column tiles per image
#define NTOT_TILES (B_*NTILES_P) // 1568

typedef __attribute__((ext_vector_type(2))) float v2f;
typedef __attribute__((ext_vector_type(8))) float v8f;
typedef __attribute__((ext_vector_type(4))) unsigned int u32x4;
typedef __attribute__((ext_vector_type(4))) int i32x4;
typedef __attribute__((ext_vector_type(8))) int i32x8;

__device__ __forceinline__ int refl(int i, int n) {
    if (i < 0)  i = -i;
    if (i >= n) i = 2 * n - 2 - i;
    return i;
}

// ------------------------------------------------------------------
// Tensor Data Mover: load a 2D tile (tile_d0 x tile_d1 f32 elements,
// row stride stride_d0 elements) from global memory into LDS.
// D# layout per CDNA5 ISA sec. 8.3/8.4:
//   group0: d0=count(1), d1=lds byte addr, d2/d3=global addr + type=2
//   group1: d0=data_size(4B), tensor dims, tile dims, dim0 stride
// Tracked by TENSORcnt; caller waits with s_wait_tensorcnt.
// ------------------------------------------------------------------
__device__ __forceinline__ void tdm_load_2d(unsigned lds_byte_off,
                                            const float* gptr,
                                            unsigned tile_d0, unsigned tile_d1,
                                            unsigned tensor_d0, unsigned tensor_d1,
                                            unsigned stride_d0) {
    unsigned long long ga = (unsigned long long)(size_t)gptr;
    u32x4 g0;
    g0[0] = 1u;                                              // count=1, user desc
    g0[1] = lds_byte_off;                                    // LDS byte address
    g0[2] = (unsigned)(ga & 0xFFFFFFFFu);                    // global_addr[31:0]
    g0[3] = (unsigned)((ga >> 32) & 0x01FFFFFFu) | (2u << 30); // addr[56:32] | type=2
    i32x8 g1;
    g1[0] = (int)(2u << 16);                                 // data_size=2 (4B)
    g1[1] = (int)((tensor_d0 & 0xFFFFu) << 16);              // tensor_dim0[15:0]
    g1[2] = (int)(((tensor_d0 >> 16) & 0xFFFFu) |
                  ((tensor_d1 & 0xFFFFu) << 16));            // dim0[31:16] | dim1[15:0]
    g1[3] = (int)(((tensor_d1 >> 16) & 0xFFFFu) |
                  ((tile_d0 & 0xFFFFu) << 16));              // dim1[31:16] | tile_dim0
    g1[4] = (int)(tile_d1 & 0xFFFFu);                        // tile_dim1 (tile_dim2=0)
    g1[5] = (int)stride_d0;                                  // tensor_dim0_stride[31:0]
    g1[6] = 0;
    g1[7] = 0;
    i32x4 z4 = {0, 0, 0, 0};
#if __clang_major__ >= 23
    i32x8 z8 = {0, 0, 0, 0, 0, 0, 0, 0};
    __builtin_amdgcn_tensor_load_to_lds(g0, g1, z4, z4, z8, 0);
#else
    __builtin_amdgcn_tensor_load_to_lds(g0, g1, z4, z4, 0);
#endif
}

// ------------------------------------------------------------------
// Kernel 1: fused projection GEMM  y[b, 0:384, p] = [w1;w2;w3] @ x[b,:,p]
// One block per 16-wide column tile.  The x panel (256x16 f32, 16 KB) is
// DMA'd into LDS once by the Tensor Data Mover; 8 waves each compute
// 3 M-tiles (48 rows) reusing one LDS B-fragment across 3 WMMAs.
// ------------------------------------------------------------------
__global__ __launch_bounds__(256)
void proj_gemm(const float* __restrict__ x,
               const float* __restrict__ w1,
               const float* __restrict__ w2,
               const float* __restrict__ w3,
               float* __restrict__ yproj) {
    __shared__ float bsh[CIN * 16];                  // [c][16] panel, 16 KB
    const int lane = threadIdx.x & 31;
    const int wave = threadIdx.x >> 5;
    const int nt   = blockIdx.x;                     // 0..1567
    const int b    = nt / NTILES_P;
    const int pbase = (nt % NTILES_P) * 16;
    const int half = lane >> 4;                      // 0: K0/K1, 1: K2/K3
    const int l15  = lane & 15;

    if (wave == 0) {
        tdm_load_2d((unsigned)(size_t)bsh,
                    x + (size_t)b * CIN * HW + pbase,
                    /*tile*/ 16, CIN, /*tensor*/ HW, CIN, /*stride*/ HW);
        __builtin_amdgcn_s_wait_tensorcnt(0);
    }
    __syncthreads();

    // 3 M-tiles per wave: rows [wave*48 .. wave*48+47]
    const float* arow[3];
#pragma unroll
    for (int t = 0; t < 3; ++t) {
        int row = (wave * 3 + t) * 16 + l15;
        if      (row < 64)   arow[t] = w1 + (size_t)row * CIN;
        else if (row < 128)  arow[t] = w2 + (size_t)(row - 64) * CIN;
        else                 arow[t] = w3 + (size_t)(row - 128) * CIN;
    }

    v8f acc0 = {}, acc1 = {}, acc2 = {};
    for (int k0 = 0; k0 < CIN; k0 += 4) {
        v2f bb;
        bb.x = bsh[(k0 + 2 * half) * 16 + l15];
        bb.y = bsh[(k0 + 2 * half + 1) * 16 + l15];
        v2f a0, a1, a2;
        a0.x = arow[0][k0 + 2 * half]; a0.y = arow[0][k0 + 2 * half + 1];
        a1.x = arow[1][k0 + 2 * half]; a1.y = arow[1][k0 + 2 * half + 1];
        a2.x = arow[2][k0 + 2 * half]; a2.y = arow[2][k0 + 2 * half + 1];
        acc0 = __builtin_amdgcn_wmma_f32_16x16x4_f32(false, a0, false, bb,
                                                     (short)0, acc0, false, false);
        acc1 = __builtin_amdgcn_wmma_f32_16x16x4_f32(false, a1, false, bb,
                                                     (short)0, acc1, false, false);
        acc2 = __builtin_amdgcn_wmma_f32_16x16x4_f32(false, a2, false, bb,
                                                     (short)0, acc2, false, false);
    }

    float* outp = yproj + (size_t)b * OC1 * HW + pbase + l15;
#pragma unroll
    for (int v = 0; v < 8; ++v) {
        outp[(size_t)((wave * 3 + 0) * 16 + v + half * 8) * HW] = acc0[v];
        outp[(size_t)((wave * 3 + 1) * 16 + v + half * 8) * HW] = acc1[v];
        outp[(size_t)((wave * 3 + 2) * 16 + v + half * 8) * HW] = acc2[v];
    }
}

// ------------------------------------------------------------------
// Kernel 2: training-mode BN stats over the 640 concat channels.
// Channels >= 64 are reflect-shifted views of y2 (rows 64..127 of yproj).
// Emits per-channel scale = rstd*g, shift = b - mean*scale.
// ------------------------------------------------------------------
__global__ __launch_bounds__(256)
void cat_stats(const float* __restrict__ yproj,
               const float* __restrict__ g1, const float* __restrict__ b1,
               float* __restrict__ scaleA, float* __restrict__ shiftA) {
    const int ch = blockIdx.x;                   // 0..639
    __shared__ float ssum[256], ssq[256];

    int src_row, di = 0, dj = 0;
    if (ch < REL) {
        src_row = ch;
    } else {
        int c2 = ch - REL;
        int r = c2 / KK, kk = c2 - r * KK;
        src_row = REL + r;
        di = kk / KSZ - 1;
        dj = kk % KSZ - 1;
    }

    float s = 0.f, q = 0.f;
    for (int idx = threadIdx.x; idx < B_ * HW; idx += 256) {
        int b = idx / HW, p = idx - b * HW;
        int i = p / W_, j = p - i * W_;
        int pp = refl(i + di, H_) * W_ + refl(j + dj, W_);
        float v = yproj[(size_t)b * OC1 * HW + (size_t)src_row * HW + pp];
        s += v; q += v * v;
    }
    ssum[threadIdx.x] = s; ssq[threadIdx.x] = q;
    __syncthreads();
    for (int off = 128; off > 0; off >>= 1) {
        if (threadIdx.x < off) {
            ssum[threadIdx.x] += ssum[threadIdx.x + off];
            ssq[threadIdx.x]  += ssq[threadIdx.x + off];
        }
        __syncthreads();
    }
    if (threadIdx.x == 0) {
        const float n = (float)(B_ * HW);
        float mean = ssum[0] / n;
        float var  = ssq[0] / n - mean * mean;
        float rstd = rsqrtf(var + EPS);
        float sc = rstd * g1[ch];
        scaleA[ch] = sc;
        shiftA[ch] = b1[ch] - mean * sc;
    }
}

// ------------------------------------------------------------------
// Kernel 3: h1 = ww1[32,640] @ relu(bn(cat)).  BN+ReLU fused into the
// B-operand gather (reflect-indexed loads of y2 for unfold channels).
// ------------------------------------------------------------------
__global__ __launch_bounds__(256)
void mlp1_gemm(const float* __restrict__ yproj,
               const float* __restrict__ ww1,
               const float* __restrict__ scaleA,
               const float* __restrict__ shiftA,
               float* __restrict__ h1) {
    const int lane = threadIdx.x & 31;
    const int wave = threadIdx.x >> 5;
    const int tile = blockIdx.x * 8 + wave;      // 2*1568 tiles exactly
    const int MT = MID / 16;                     // 2
    const int mt = tile % MT;
    const int nt = tile / MT;
    const int b  = nt / NTILES_P;
    const int pbase = (nt % NTILES_P) * 16;
    const int half = lane >> 4;
    const int l15  = lane & 15;
    const int p = pbase + l15;
    const int i = p / W_, j = p - i * W_;

    const float* arow = ww1 + (size_t)(mt * 16 + l15) * CCAT;
    const float* ybase = yproj + (size_t)b * OC1 * HW;

    v8f acc = {};
    for (int k0 = 0; k0 < CCAT; k0 += 4) {
        v2f a, bb;
        a.x = arow[k0 + 2 * half];
        a.y = arow[k0 + 2 * half + 1];
        const int ch0 = k0 + 2 * half;
#pragma unroll
        for (int e = 0; e < 2; ++e) {
            int ch = ch0 + e;
            float v;
            if (ch < REL) {
                v = ybase[(size_t)ch * HW + p];
            } else {
                int c2 = ch - REL;
                int r = c2 / KK, kk = c2 - r * KK;
                int pp = refl(i + kk / KSZ - 1, H_) * W_ +
                         refl(j + kk % KSZ - 1, W_);
                v = ybase[(size_t)(REL + r) * HW + pp];
            }
            v = fmaxf(v * scaleA[ch] + shiftA[ch], 0.f);
            if (e == 0) bb.x = v; else bb.y = v;
        }
        acc = __builtin_amdgcn_wmma_f32_16x16x4_f32(false, a, false, bb,
                                                    (short)0, acc, false, false);
    }

    float* outp = h1 + (size_t)b * MID * HW + p;
#pragma unroll
    for (int v = 0; v < 8; ++v)
        outp[(size_t)(mt * 16 + v + half * 8) * HW] = acc[v];
}

// ------------------------------------------------------------------
// Kernel 4: BN stats for h1 (32 channels).
// ------------------------------------------------------------------
__global__ __launch_bounds__(256)
void h1_stats(const float* __restrict__ h1,
              const float* __restrict__ g2, const float* __restrict__ b2,
              float* __restrict__ scaleB, float* __restrict__ shiftB) {
    const int ch = blockIdx.x;                   // 0..31
    __shared__ float ssum[256], ssq[256];
    float s = 0.f, q = 0.f;
    for (int idx = threadIdx.x; idx < B_ * HW; idx += 256) {
        int b = idx / HW, p = idx - b * HW;
        float v = h1[(size_t)b * MID * HW + (size_t)ch * HW + p];
        s += v; q += v * v;
    }
    ssum[threadIdx.x] = s; ssq[threadIdx.x] = q;
    __syncthreads();
    for (int off = 128; off > 0; off >>= 1) {
        if (threadIdx.x < off) {
            ssum[threadIdx.x] += ssum[threadIdx.x + off];
            ssq[threadIdx.x]  += ssq[threadIdx.x + off];
        }
        __syncthreads();
    }
    if (threadIdx.x == 0) {
        const float n = (float)(B_ * HW);
        float mean = ssum[0] / n;
        float var  = ssq[0] / n - mean * mean;
        float rstd = rsqrtf(var + EPS);
        float sc = rstd * g2[ch];
        scaleB[ch] = sc;
        shiftB[ch] = b2[ch] - mean * sc;
    }
}

// ------------------------------------------------------------------
// Kernel 5: wgt = ww2[288,32] @ relu(bn(h1)).
// One block per column tile; h1 panel (32x16 f32, 2 KB) staged to LDS by
// the TDM; 9 waves x 2 M-tiles = 18 M-tiles; K=32 -> 8 WMMAs per tile.
// ------------------------------------------------------------------
__global__ __launch_bounds__(288)
void mlp2_gemm(const float* __restrict__ h1,
               const float* __restrict__ ww2,
               const float* __restrict__ scaleB,
               const float* __restrict__ shiftB,
               float* __restrict__ wgt) {
    __shared__ float hsh[MID * 16];              // [c][16] panel, 2 KB
    const int lane = threadIdx.x & 31;
    const int wave = threadIdx.x >> 5;           // 0..8
    const int nt   = blockIdx.x;                 // 0..1567
    const int b    = nt / NTILES_P;
    const int pbase = (nt % NTILES_P) * 16;
    const int half = lane >> 4;
    const int l15  = lane & 15;

    if (wave == 0) {
        tdm_load_2d((unsigned)(size_t)hsh,
                    h1 + (size_t)b * MID * HW + pbase,
                    /*tile*/ 16, MID, /*tensor*/ HW, MID, /*stride*/ HW);
        __builtin_amdgcn_s_wait_tensorcnt(0);
    }
    __syncthreads();

    const float* arow0 = ww2 + (size_t)((wave * 2 + 0) * 16 + l15) * MID;
    const float* arow1 = ww2 + (size_t)((wave * 2 + 1) * 16 + l15) * MID;

    v8f acc0 = {}, acc1 = {};
#pragma unroll
    for (int k0 = 0; k0 < MID; k0 += 4) {
        const int ch = k0 + 2 * half;
        v2f bb;
        bb.x = fmaxf(hsh[ch * 16 + l15] * scaleB[ch] + shiftB[ch], 0.f);
        bb.y = fmaxf(hsh[(ch + 1) * 16 + l15] * scaleB[ch + 1] + shiftB[ch + 1], 0.f);
        v2f a0, a1;
        a0.x = arow0[ch]; a0.y = arow0[ch + 1];
        a1.x = arow1[ch]; a1.y = arow1[ch + 1];
        acc0 = __builtin_amdgcn_wmma_f32_16x16x4_f32(false, a0, false, bb,
                                                     (short)0, acc0, false, false);
        acc1 = __builtin_amdgcn_wmma_f32_16x16x4_f32(false, a1, false, bb,
                                                     (short)0, acc1, false, false);
    }

    float* outp = wgt + (size_t)b * (MID * KK) * HW + pbase + l15;
#pragma unroll
    for (int v = 0; v < 8; ++v) {
        outp[(size_t)((wave * 2 + 0) * 16 + v + half * 8) * HW] = acc0[v];
        outp[(size_t)((wave * 2 + 1) * 16 + v + half * 8) * HW] = acc1[v];
    }
}

// ------------------------------------------------------------------
// Kernel 6: out[b, m*8+s, p] = sum_kk x3[b, m*8+s, refl-nbr(p,kk)] *
//                                      wgt[b, m*9+kk, p]
// x3 lives at rows 128..383 of yproj.  Pure streaming kernel.
// ------------------------------------------------------------------
__global__ __launch_bounds__(256)
void aggregate(const float* __restrict__ yproj,
               const float* __restrict__ wgt,
               float* __restrict__ out) {
    size_t gid = (size_t)blockIdx.x * 256 + threadIdx.x;
    if (gid >= (size_t)B_ * OUTC * HW) return;
    int p  = (int)(gid % HW);
    int t  = (int)(gid / HW);
    int oc = t % OUTC;
    int b  = t / OUTC;
    int m  = oc >> 3;                            // / SHARE
    int i = p / W_, j = p - i * W_;

    const float* x3 = yproj + (size_t)b * OC1 * HW + (size_t)(128 + oc) * HW;
    const float* wg = wgt + (size_t)b * (MID * KK) * HW + (size_t)(m * KK) * HW + p;

    float acc = 0.f;
#pragma unroll
    for (int kk = 0; kk < KK; ++kk) {
        int pp = refl(i + kk / KSZ - 1, H_) * W_ + refl(j + kk % KSZ - 1, W_);
        acc += x3[pp] * wg[(size_t)kk * HW];
    }
    out[gid] = acc;
}

// ------------------------------------------------------------------
extern "C" void kernel_launch(void* const* d_in, const int* in_sizes, int n_in,
                              void* d_out, int out_size, void* d_ws, size_t ws_size,
                              hipStream_t stream) {
    const float* x   = (const float*)d_in[0];
    const float* w1  = (const float*)d_in[1];
    const float* w2  = (const float*)d_in[2];
    const float* w3  = (const float*)d_in[3];
    const float* g1  = (const float*)d_in[4];
    const float* b1  = (const float*)d_in[5];
    const float* ww1 = (const float*)d_in[6];
    const float* g2  = (const float*)d_in[7];
    const float* b2  = (const float*)d_in[8];
    const float* ww2 = (const float*)d_in[9];

    float* ws     = (float*)d_ws;
    float* yproj  = ws;                                   // 8*384*3136
    float* h1     = yproj + (size_t)B_ * OC1 * HW;        // 8*32*3136
    float* wgt    = h1    + (size_t)B_ * MID * HW;        // 8*288*3136
    float* scaleA = wgt   + (size_t)B_ * MID * KK * HW;   // 640
    float* shiftA = scaleA + CCAT;                        // 640
    float* scaleB = shiftA + CCAT;                        // 32
    float* shiftB = scaleB + MID;                         // 32
    float* out    = (float*)d_out;

    proj_gemm<<<NTOT_TILES, 256, 0, stream>>>(x, w1, w2, w3, yproj);
    cat_stats<<<CCAT, 256, 0, stream>>>(yproj, g1, b1, scaleA, shiftA);
    mlp1_gemm<<<(2 * NTOT_TILES) / 8, 256, 0, stream>>>(yproj, ww1, scaleA, shiftA, h1);
    h1_stats<<<MID, 256, 0, stream>>>(h1, g2, b2, scaleB, shiftB);
    mlp2_gemm<<<NTOT_TILES, 288, 0, stream>>>(h1, ww2, scaleB, shiftB, wgt);

    const size_t nout = (size_t)B_ * OUTC * HW;
    aggregate<<<(unsigned)((nout + 255) / 256), 256, 0, stream>>>(yproj, wgt, out);
}